// WinCorrTorch_81647328297400
// MI455X (gfx1250) — compile-verified
//
#include <hip/hip_runtime.h>
#include <hip/hip_bf16.h>
#include <stdint.h>

// ---------------- problem constants ----------------
#define BATCH 2
#define CH    64
#define DD    80
#define HH    80
#define WW    80
#define PLANE (HH * WW)        // 6400
#define VOL   (DD * HH * WW)   // 512000
#define NOFF  27

// ---------------- tiling ----------------
#define TZ 8
#define TH 8
#define TWIDE 8                // w extent of tile (4 thread-pairs * 2)
#define HALO 10                // TZ+2
#define KC 8                   // channels per LDS block
#define NB (CH / KC)           // 8 channel blocks
#define HALO_ELEMS (KC * HALO * HALO * HALO)  // 8000 floats per block

typedef unsigned int u32;
typedef __attribute__((ext_vector_type(4))) u32 u32x4;
typedef __attribute__((ext_vector_type(4))) int i32x4;
typedef __attribute__((ext_vector_type(8))) int i32x8;
typedef __attribute__((ext_vector_type(2))) float f32x2;  // native vector (HIP float2 is a class)

#if __has_builtin(__builtin_amdgcn_global_load_async_to_lds_b32)
#define HAVE_ASYNC 1
#else
#define HAVE_ASYNC 0
#endif

#if __has_builtin(__builtin_amdgcn_tensor_load_to_lds) && \
    __has_builtin(__builtin_amdgcn_s_wait_tensorcnt)
#define HAVE_TDM 1
#else
#define HAVE_TDM 0
#endif

// ---- TDM: one 4-D descriptor copies a (10 x 10 x 10 x KC) halo tile
//      (x,y,z,c order; x fastest) from global y into packed LDS.
__device__ __forceinline__ void tdm_load_tile(u32 ldsByteAddr, const float* gsrc) {
#if HAVE_TDM
  unsigned long long ga = (unsigned long long)(size_t)gsrc;
  // D# group 0: count=1(valid), lds_addr, 57-bit global addr, type=2 ("image")
  u32x4 g0;
  g0[0] = 1u;
  g0[1] = ldsByteAddr;
  g0[2] = (u32)(ga & 0xFFFFFFFFu);
  g0[3] = (u32)((ga >> 32) & 0x01FFFFFFu) | (2u << 30);
  // D# group 1
  i32x8 g1;
  g1[0] = (int)(2u << 16);                    // workgroup_mask=0, data_size=2 (4B)
  g1[1] = (int)((WW & 0xFFFF) << 16);         // tensor_dim0[15:0] in bits[63:48]
  g1[2] = (int)((HH & 0xFFFF) << 16);         // dim0 hi16=0 | tensor_dim1 lo16
  g1[3] = (int)(HALO << 16);                  // dim1 hi16=0 | tile_dim0=10
  g1[4] = (int)(HALO | (HALO << 16));         // tile_dim1=10 | tile_dim2=10
  g1[5] = (int)WW;                            // tensor_dim0_stride lo32 = 80 (row pitch)
  g1[6] = (int)((PLANE & 0xFFFF) << 16);      // dim0_stride hi16=0 | dim1_stride lo16 (6400)
  g1[7] = (int)(PLANE >> 16);                 // dim1_stride bits[47:16] = 0
  // D# group 2 (normal mode)
  i32x4 g2;
  g2[0] = DD;                                 // tensor_dim2
  g2[1] = CH;                                 // tensor_dim3
  g2[2] = VOL;                                // tensor_dim2_stride lo32 = 512000 (channel pitch)
  g2[3] = (int)(KC << 16);                    // dim2_stride hi16=0 | tile_dim3=KC
  i32x4 g3 = {0, 0, 0, 0};                    // no dim4
  i32x8 g4 = {0, 0, 0, 0, 0, 0, 0, 0};        // extra group (clang-23 6-arg form)
  __builtin_amdgcn_tensor_load_to_lds(g0, g1, g2, g3, g4, 0);
#else
  (void)ldsByteAddr; (void)gsrc;
#endif
}

__device__ __forceinline__ void wait_stage() {
#if HAVE_TDM
  __builtin_amdgcn_s_wait_tensorcnt(0);
#endif
#if HAVE_ASYNC
#if __has_builtin(__builtin_amdgcn_s_wait_asynccnt)
  __builtin_amdgcn_s_wait_asynccnt(0);
#else
  asm volatile("s_wait_asynccnt 0x0" ::: "memory");
#endif
#endif
}

// Stage one KC-channel halo block into LDS buffer `buf`.
__device__ __forceinline__ void stage_block(float* buf, const float* ybatch, int cb,
                                            int z0, int h0, int w0t, int tid,
                                            bool interior) {
  if (HAVE_TDM && interior) {
    if (tid < 32) {  // one wave issues the DMA (EXEC ignored by tensor ops)
      const float* gsrc = ybatch + (size_t)cb * VOL +
                          (size_t)(z0 - 1) * PLANE + (size_t)(h0 - 1) * WW + (w0t - 1);
      tdm_load_tile((u32)(size_t)buf, gsrc);
    }
  } else {
    // generic path: per-element predicated async copy; OOB slots stay zero
    for (int i = tid; i < HALO_ELEMS; i += 256) {
      int kc = i / 1000;
      int r  = i - kc * 1000;
      int hz = r / 100;  r -= hz * 100;
      int hy = r / 10;
      int hx = r - hy * 10;
      int gz = z0 - 1 + hz, gy = h0 - 1 + hy, gx = w0t - 1 + hx;
      if ((u32)gz < DD && (u32)gy < HH && (u32)gx < WW) {
        const float* g = ybatch + (size_t)(cb + kc) * VOL +
                         (size_t)gz * PLANE + (size_t)gy * WW + gx;
        float* l = buf + i;
#if HAVE_ASYNC
        __builtin_amdgcn_global_load_async_to_lds_b32(
            (__attribute__((address_space(1))) int*)(size_t)g,
            (__attribute__((address_space(3))) int*)(u32)(size_t)l, 0, 0);
#else
        *l = *g;
#endif
      }
    }
  }
}

__global__ __launch_bounds__(256) void wincorr_kernel(const float* __restrict__ x,
                                                      const float* __restrict__ y,
                                                      float* __restrict__ out) {
  __shared__ float ybuf[2][HALO_ELEMS];  // 64 KB double-buffered halo tiles

  const int tid = threadIdx.x;
  const int wt = blockIdx.x, ht = blockIdx.y;
  const int zb = blockIdx.z;
  const int b  = zb / 10, zt = zb - b * 10;
  const int z0 = zt * TZ, h0 = ht * TH, w0t = wt * TWIDE;

  const int tz  = tid >> 5;         // 0..7
  const int thh = (tid >> 2) & 7;   // 0..7
  const int tw  = tid & 3;          // 0..3 (pair of w positions)
  const int z = z0 + tz, h = h0 + thh, w = w0t + tw * 2;

  const bool interior = (z0 != 0) && (z0 != DD - TZ) &&
                        (h0 != 0) && (h0 != HH - TH) &&
                        (w0t != 0) && (w0t != WW - TWIDE);

  if (!interior) {  // boundary tile: out-of-range halo slots must read as 0
    for (int i = tid; i < 2 * HALO_ELEMS; i += 256) ((float*)ybuf)[i] = 0.0f;
  }
  __syncthreads();

  const float* ybatch = y + (size_t)b * CH * VOL;
  const float* xbatch = x + (size_t)b * CH * VOL;

  stage_block(ybuf[0], ybatch, 0, z0, h0, w0t, tid, interior);
  wait_stage();
  __syncthreads();

  float acc[NOFF][2];
#pragma unroll
  for (int o = 0; o < NOFF; ++o) { acc[o][0] = 0.0f; acc[o][1] = 0.0f; }

#pragma unroll 1
  for (int kb = 0; kb < NB; ++kb) {
    const float* buf = ybuf[kb & 1];
    if (kb + 1 < NB)
      stage_block(ybuf[(kb + 1) & 1], ybatch, (kb + 1) * KC, z0, h0, w0t, tid, interior);

#pragma unroll
    for (int kc = 0; kc < KC; ++kc) {
      const int c = kb * KC + kc;
      // x is consumed exactly once per element in the whole kernel ->
      // non-temporal load keeps the 262 MB x stream from evicting y in L2.
      const f32x2 xv = __builtin_nontemporal_load(
          (const f32x2*)(xbatch + (size_t)c * VOL +
                         (size_t)z * PLANE + (size_t)h * WW + w));
      if (kb + 1 < NB)  // pull next block's x line toward the caches
        __builtin_prefetch(xbatch + (size_t)(c + KC) * VOL +
                           (size_t)z * PLANE + (size_t)h * WW + w, 0, 0);
#pragma unroll
      for (int dz = 0; dz < 3; ++dz) {
#pragma unroll
        for (int dy = 0; dy < 3; ++dy) {
          const float* r = buf + (((kc * HALO + (tz + dz)) * HALO + (thh + dy)) * HALO + tw * 2);
          const f32x2 ya = *(const f32x2*)(r);      // halo x: tw*2 .. tw*2+1
          const f32x2 yb = *(const f32x2*)(r + 2);  // halo x: tw*2+2 .. tw*2+3
          const int o = (dz * 3 + dy) * 3;
          // dx = 0,1,2  ->  shift -1,0,+1 ; positions w (x0) and w+1 (x1)
          acc[o + 0][0] = fmaf(xv.x, ya.x, acc[o + 0][0]);
          acc[o + 0][1] = fmaf(xv.y, ya.y, acc[o + 0][1]);
          acc[o + 1][0] = fmaf(xv.x, ya.y, acc[o + 1][0]);
          acc[o + 1][1] = fmaf(xv.y, yb.x, acc[o + 1][1]);
          acc[o + 2][0] = fmaf(xv.x, yb.x, acc[o + 2][0]);
          acc[o + 2][1] = fmaf(xv.y, yb.y, acc[o + 2][1]);
        }
      }
    }
    wait_stage();
    __syncthreads();
  }

  const float scale = 0.125f;  // 64^(-1/2)
  float* obase = out + (size_t)b * NOFF * VOL + (size_t)z * PLANE + (size_t)h * WW + w;
#pragma unroll
  for (int o = 0; o < NOFF; ++o) {
    f32x2 v;
    v.x = acc[o][0] * scale;
    v.y = acc[o][1] * scale;
    // write-once output: non-temporal store, don't pollute L2
    __builtin_nontemporal_store(v, (f32x2*)(obase + (size_t)o * VOL));
  }
}

extern "C" void kernel_launch(void* const* d_in, const int* in_sizes, int n_in,
                              void* d_out, int out_size, void* d_ws, size_t ws_size,
                              hipStream_t stream) {
  (void)in_sizes; (void)n_in; (void)out_size; (void)d_ws; (void)ws_size;
  const float* x = (const float*)d_in[0];
  const float* y = (const float*)d_in[1];
  float* out = (float*)d_out;
  dim3 grid(WW / TWIDE, HH / TH, BATCH * (DD / TZ));  // (10, 10, 20)
  wincorr_kernel<<<grid, dim3(256), 0, stream>>>(x, y, out);
}